// Pct_Rotation_1133871366466
// MI455X (gfx1250) — compile-verified
//
#include <hip/hip_runtime.h>
#include <math.h>

typedef float v2f __attribute__((ext_vector_type(2)));
typedef float v8f __attribute__((ext_vector_type(8)));

#define TILE_M 64
#define TILE_N 64
#define TILE_K 16

// ---------------------------------------------------------------------------
// Tiled fp32 WMMA GEMM: C[b] = A[b] (MxK) * B[b] (KxN)  (+ bias[m])
// Optional per-K row affine + ReLU applied to B while staging to LDS
// (used to fuse BatchNorm+ReLU of the previous layer into the next GEMM).
// 128 threads = 4 waves; each wave computes a 32x32 quadrant as 2x2 WMMA tiles.
// ---------------------------------------------------------------------------
template <bool TRANS_A>
__global__ __launch_bounds__(128) void gemm_wmma(
    const float* __restrict__ A, const float* __restrict__ B,
    float* __restrict__ C, int M, int N, int K, int lda, int ldb, int ldc,
    long sA, long sB, long sC,
    const float* __restrict__ bias,
    const float* __restrict__ bScale, const float* __restrict__ bShift) {
  __shared__ float As[TILE_M][TILE_K + 1];
  __shared__ float Bs[TILE_K][TILE_N + 1];
  const float* Ab = A + (long)blockIdx.z * sA;
  const float* Bb = B + (long)blockIdx.z * sB;
  float* Cb = C + (long)blockIdx.z * sC;
  int m0 = blockIdx.y * TILE_M, n0 = blockIdx.x * TILE_N;
  int lane = threadIdx.x & 31;
  int wave = threadIdx.x >> 5;
  int wr = (wave >> 1) * 32, wc = (wave & 1) * 32;
  int lm = lane & 15;
  int lk = (lane >> 4) * 2;  // K pair base: lanes 0-15 -> K0/K1, 16-31 -> K2/K3
  v8f acc00 = {}, acc01 = {}, acc10 = {}, acc11 = {};
  for (int k0 = 0; k0 < K; k0 += TILE_K) {
    if (threadIdx.x == 0 && k0 + TILE_K < K) {
      // emits global_prefetch_b8 for the next K block
      __builtin_prefetch(TRANS_A ? &Ab[(long)(k0 + TILE_K) * lda + m0]
                                 : &Ab[(long)m0 * lda + k0 + TILE_K], 0, 0);
      __builtin_prefetch(&Bb[(long)(k0 + TILE_K) * ldb + n0], 0, 0);
    }
    for (int e = threadIdx.x; e < TILE_M * TILE_K; e += 128) {
      int m, k;
      if (TRANS_A) { k = e >> 6; m = e & 63; } else { m = e >> 4; k = e & 15; }
      int gm = m0 + m, gk = k0 + k;
      float v = 0.f;
      if (gm < M && gk < K)
        v = TRANS_A ? Ab[(long)gk * lda + gm] : Ab[(long)gm * lda + gk];
      As[m][k] = v;
    }
    for (int e = threadIdx.x; e < TILE_K * TILE_N; e += 128) {
      int k = e >> 6, n = e & 63;
      int gk = k0 + k, gn = n0 + n;
      float v = 0.f;
      if (gk < K && gn < N) {
        v = Bb[(long)gk * ldb + gn];
        if (bScale) v = fmaxf(v * bScale[gk] + bShift[gk], 0.f);
      }
      Bs[k][n] = v;
    }
    __syncthreads();
#pragma unroll
    for (int kk = 0; kk < TILE_K; kk += 4) {
      v2f a0, a1, b0, b1;
      a0.x = As[wr + lm][kk + lk];      a0.y = As[wr + lm][kk + lk + 1];
      a1.x = As[wr + 16 + lm][kk + lk]; a1.y = As[wr + 16 + lm][kk + lk + 1];
      b0.x = Bs[kk + lk][wc + lm];      b0.y = Bs[kk + lk + 1][wc + lm];
      b1.x = Bs[kk + lk][wc + 16 + lm]; b1.y = Bs[kk + lk + 1][wc + 16 + lm];
      acc00 = __builtin_amdgcn_wmma_f32_16x16x4_f32(false, a0, false, b0, (short)0, acc00, false, false);
      acc01 = __builtin_amdgcn_wmma_f32_16x16x4_f32(false, a0, false, b1, (short)0, acc01, false, false);
      acc10 = __builtin_amdgcn_wmma_f32_16x16x4_f32(false, a1, false, b0, (short)0, acc10, false, false);
      acc11 = __builtin_amdgcn_wmma_f32_16x16x4_f32(false, a1, false, b1, (short)0, acc11, false, false);
    }
    __syncthreads();
  }
  // C/D layout: VGPR r -> M = r (lanes 0-15) / r+8 (lanes 16-31), N = lane&15
  int mrow = (lane >> 4) << 3;
#pragma unroll
  for (int r = 0; r < 8; ++r) {
    int mA = m0 + wr + mrow + r;
    int mB = mA + 16;
    int nA = n0 + wc + lm;
    int nB = nA + 16;
    if (mA < M) {
      float bi = bias ? bias[mA] : 0.f;
      if (nA < N) Cb[(long)mA * ldc + nA] = acc00[r] + bi;
      if (nB < N) Cb[(long)mA * ldc + nB] = acc01[r] + bi;
    }
    if (mB < M) {
      float bi = bias ? bias[mB] : 0.f;
      if (nA < N) Cb[(long)mB * ldc + nA] = acc10[r] + bi;
      if (nB < N) Cb[(long)mB * ldc + nB] = acc11[r] + bi;
    }
  }
}

// ---------------------------------------------------------------------------
// BatchNorm statistics: per-channel mean/var over (B, N) -> scale/shift
// Layout: row (b,c) at x + b*bstride + c*N.
// ---------------------------------------------------------------------------
__global__ __launch_bounds__(256) void stats_kernel(
    const float* __restrict__ x, int Bn, int C, long N, long bstride,
    const float* __restrict__ g, const float* __restrict__ bta,
    float* __restrict__ scale, float* __restrict__ shift) {
  __shared__ double sh_s[256];
  __shared__ double sh_q[256];
  int c = blockIdx.x;
  int t = threadIdx.x;
  double s = 0.0, q = 0.0;
  for (int b = 0; b < Bn; ++b) {
    const float* row = x + (long)b * bstride + (long)c * N;
    for (long i = t; i < N; i += 256) { double v = row[i]; s += v; q += v * v; }
  }
  sh_s[t] = s; sh_q[t] = q;
  __syncthreads();
  for (int o = 128; o > 0; o >>= 1) {
    if (t < o) { sh_s[t] += sh_s[t + o]; sh_q[t] += sh_q[t + o]; }
    __syncthreads();
  }
  if (t == 0) {
    double cnt = (double)Bn * (double)N;
    double m = sh_s[0] / cnt;
    double var = sh_q[0] / cnt - m * m;
    float sc = (float)((double)g[c] / sqrt(var + 1e-3));
    scale[c] = sc;
    shift[c] = bta[c] - (float)m * sc;
  }
}

// out = [res +] act(in*scale[c] + shift[c]);  act: 0=relu 1=leaky0.2 2=none
__global__ void ew_affine(const float* __restrict__ in, long inStride,
                          const float* __restrict__ res, long resStride,
                          float* __restrict__ out, long outStride,
                          const float* __restrict__ scale,
                          const float* __restrict__ shift,
                          int C, int N, long total, int act) {
  long idx = (long)blockIdx.x * blockDim.x + threadIdx.x;
  if (idx >= total) return;
  long per = (long)C * N;
  long b = idx / per;
  long rem = idx - b * per;
  int c = (int)(rem / N);
  float v = in[b * inStride + rem] * scale[c] + shift[c];
  if (act == 0) v = fmaxf(v, 0.f);
  else if (act == 1) v = v > 0.f ? v : 0.2f * v;
  if (res) v += res[b * resStride + rem];
  out[b * outStride + rem] = v;
}

__global__ void ew_sub(const float* __restrict__ a, long aStride,
                       const float* __restrict__ b, long bStride,
                       float* __restrict__ o, long per, long total) {
  long idx = (long)blockIdx.x * blockDim.x + threadIdx.x;
  if (idx >= total) return;
  long bb = idx / per;
  long rem = idx - bb * per;
  o[idx] = a[bb * aStride + rem] - b[bb * bStride + rem];
}

__global__ void transpose_xyz(const float* __restrict__ x, float* __restrict__ xyz,
                              int Bn, int N) {
  int idx = blockIdx.x * blockDim.x + threadIdx.x;
  if (idx >= Bn * N) return;
  int b = idx / N, n = idx - b * N;
  const float* xb = x + (long)b * 3 * N;
  float* o = xyz + (long)idx * 3;
  o[0] = xb[n]; o[1] = xb[N + n]; o[2] = xb[2 * N + n];
}

// Farthest point sampling; deterministic start at index 0, argmax ties -> lowest idx.
__global__ __launch_bounds__(256) void fps_kernel(const float* __restrict__ pts,
                                                  int N, int S,
                                                  int* __restrict__ fidx,
                                                  float* __restrict__ newxyz) {
  __shared__ float dist[1024];
  __shared__ float rv[256];
  __shared__ int ri[256];
  __shared__ int curFar;
  int b = blockIdx.x, t = threadIdx.x;
  const float* p = pts + (long)b * N * 3;
  for (int n = t; n < N; n += 256) dist[n] = 1e10f;
  if (t == 0) curFar = 0;
  __syncthreads();
  for (int it = 0; it < S; ++it) {
    int far = curFar;
    float fx = p[far * 3 + 0], fy = p[far * 3 + 1], fz = p[far * 3 + 2];
    if (t == 0) {
      fidx[b * S + it] = far;
      float* o = newxyz + ((long)b * S + it) * 3;
      o[0] = fx; o[1] = fy; o[2] = fz;
    }
    float bv = -1.f; int bi = N;
    for (int n = t; n < N; n += 256) {
      float dx = p[n * 3] - fx, dy = p[n * 3 + 1] - fy, dz = p[n * 3 + 2] - fz;
      float d = dx * dx + dy * dy + dz * dz;
      float dd = fminf(dist[n], d);
      dist[n] = dd;
      if (dd > bv) { bv = dd; bi = n; }
    }
    rv[t] = bv; ri[t] = bi;
    __syncthreads();
    for (int o = 128; o > 0; o >>= 1) {
      if (t < o) {
        if (rv[t + o] > rv[t] || (rv[t + o] == rv[t] && ri[t + o] < ri[t])) {
          rv[t] = rv[t + o]; ri[t] = ri[t + o];
        }
      }
      __syncthreads();
    }
    if (t == 0) curFar = ri[0];
    __syncthreads();
  }
}

// kNN: 32 smallest distances (selection by repeated argmin; set equality is what
// matters downstream because local_op max-pools over neighbors).
__global__ __launch_bounds__(128) void knn_kernel(const float* __restrict__ pts,
                                                  const float* __restrict__ ctr,
                                                  int N, int S, int Kn,
                                                  int* __restrict__ nidx) {
  __shared__ float d[1024];
  __shared__ float rv[128];
  __shared__ int ri[128];
  int s = blockIdx.x, b = blockIdx.y, t = threadIdx.x;
  const float* p = pts + (long)b * N * 3;
  const float* c = ctr + ((long)b * S + s) * 3;
  float cx = c[0], cy = c[1], cz = c[2];
  for (int n = t; n < N; n += 128) {
    float dx = p[n * 3] - cx, dy = p[n * 3 + 1] - cy, dz = p[n * 3 + 2] - cz;
    d[n] = dx * dx + dy * dy + dz * dz;
  }
  __syncthreads();
  int* outp = nidx + ((long)b * S + s) * Kn;
  for (int j = 0; j < Kn; ++j) {
    float bv = 3.4e38f; int bi = N;
    for (int n = t; n < N; n += 128)
      if (d[n] < bv) { bv = d[n]; bi = n; }
    rv[t] = bv; ri[t] = bi;
    __syncthreads();
    for (int o = 64; o > 0; o >>= 1) {
      if (t < o) {
        if (rv[t + o] < rv[t] || (rv[t + o] == rv[t] && ri[t + o] < ri[t])) {
          rv[t] = rv[t + o]; ri[t] = ri[t + o];
        }
      }
      __syncthreads();
    }
    if (t == 0) { outp[j] = ri[0]; d[ri[0]] = 3.4e38f; }
    __syncthreads();
  }
}

// Build grouped features feat(c,g) = concat(P[nb]-P[center], P[center]) directly
// in (2D x G) column layout (G = B*S*K, K contiguous per group).
__global__ void gather_feat(const float* __restrict__ P, const int* __restrict__ nidx,
                            const int* __restrict__ fidx, float* __restrict__ feat,
                            int Bn, int D, int Np, int S, int Kn) {
  long G = (long)Bn * S * Kn;
  long g = (long)blockIdx.x * blockDim.x + threadIdx.x;
  int c = blockIdx.y;
  if (g >= G) return;
  int k = (int)(g % Kn);
  long bs = g / Kn;
  int n = (int)(bs % S);
  int b = (int)(bs / S);
  int nb = nidx[((long)b * S + n) * Kn + k];
  int ce = fidx[b * S + n];
  const float* Pb = P + (long)b * D * Np;
  float v;
  if (c < D) v = Pb[(long)c * Np + nb] - Pb[(long)c * Np + ce];
  else       v = Pb[(long)(c - D) * Np + ce];
  feat[(long)c * G + g] = v;
}

// out[b, c, s] = max_k relu(in[c, (b*S+s)*K + k] * scale[c] + shift[c])
__global__ void maxpool_local(const float* __restrict__ in,
                              const float* __restrict__ scale,
                              const float* __restrict__ shift,
                              float* __restrict__ out, long outBStride,
                              int Bn, int C, int S, int Kn) {
  long idx = (long)blockIdx.x * blockDim.x + threadIdx.x;
  long total = (long)Bn * C * S;
  if (idx >= total) return;
  int s = (int)(idx % S);
  long r = idx / S;
  int c = (int)(r % C);
  int b = (int)(r / C);
  long G = (long)Bn * S * Kn;
  const float* row = in + (long)c * G + ((long)b * S + s) * Kn;
  float sc = scale[c], sf = shift[c];
  float m = -3.4e38f;
  for (int k = 0; k < Kn; ++k) m = fmaxf(m, fmaxf(row[k] * sc + sf, 0.f));
  out[(long)b * outBStride + (long)c * S + s] = m;
}

__global__ __launch_bounds__(256) void softmax_rows(float* __restrict__ att, int Nn) {
  __shared__ float red[256];
  long row = blockIdx.x;
  float* r = att + row * (long)Nn;
  int t = threadIdx.x;
  float v = (t < Nn) ? r[t] : -3.4e38f;
  red[t] = v;
  __syncthreads();
  for (int o = 128; o > 0; o >>= 1) { if (t < o) red[t] = fmaxf(red[t], red[t + o]); __syncthreads(); }
  float mx = red[0];
  __syncthreads();
  float e = (t < Nn) ? expf(v - mx) : 0.f;
  red[t] = e;
  __syncthreads();
  for (int o = 128; o > 0; o >>= 1) { if (t < o) red[t] += red[t + o]; __syncthreads(); }
  if (t < Nn) r[t] = e / red[0];
}

__global__ void colsum_kernel(const float* __restrict__ att, float* __restrict__ cs,
                              int Nn, int total) {
  int idx = blockIdx.x * blockDim.x + threadIdx.x;
  if (idx >= total) return;
  int b = idx / Nn, m = idx - b * Nn;
  const float* base = att + (long)b * Nn * Nn + m;
  float s = 0.f;
  for (int n = 0; n < Nn; ++n) s += base[(long)n * Nn];
  cs[idx] = s;
}

__global__ void renorm_kernel(float* __restrict__ att, const float* __restrict__ cs,
                              int Nn, long total) {
  long idx = (long)blockIdx.x * blockDim.x + threadIdx.x;
  if (idx >= total) return;
  long per = (long)Nn * Nn;
  long b = idx / per;
  int m = (int)(idx % Nn);
  att[idx] = att[idx] / (1e-9f + cs[b * Nn + m]);
}

// fvT[c, b] = max_n leaky0.2(zz[b,c,n]*sc[c]+sh[c])  (transposed for next GEMM)
__global__ void maxpool_leaky_T(const float* __restrict__ zz,
                                const float* __restrict__ sc,
                                const float* __restrict__ sh,
                                float* __restrict__ fvT, int Bn, int C, int Nn) {
  int idx = blockIdx.x * blockDim.x + threadIdx.x;
  if (idx >= C * Bn) return;
  int b = idx % Bn, c = idx / Bn;
  const float* r = zz + ((long)b * C + c) * Nn;
  float s = sc[c], f = sh[c];
  float m = -3.4e38f;
  for (int n = 0; n < Nn; ++n) {
    float v = r[n] * s + f;
    v = v > 0.f ? v : 0.2f * v;
    m = fmaxf(m, v);
  }
  fvT[(long)c * Bn + b] = m;
}

__global__ void final_store(const float* __restrict__ o3, float* __restrict__ out,
                            int Bn, int M) {
  int idx = blockIdx.x * blockDim.x + threadIdx.x;
  if (idx >= Bn * M) return;
  int o = idx % M, b = idx / M;
  out[b * M + o] = o3[(long)o * Bn + b];
}

// ---------------------------------------------------------------------------
extern "C" void kernel_launch(void* const* d_in, const int* in_sizes, int n_in,
                              void* d_out, int out_size, void* d_ws, size_t ws_size,
                              hipStream_t stream) {
  (void)in_sizes; (void)n_in; (void)out_size; (void)ws_size;
  const float* X = (const float*)d_in[0];
  auto P = [&](int i) { return (const float*)d_in[i]; };

  char* ws = (char*)d_ws;
  size_t cur = 0;
  auto alloc = [&](size_t floats) {
    size_t off = cur;
    cur += ((floats * 4 + 255) / 256) * 256;
    return (float*)(ws + off);
  };
  float* bufA = alloc(67108864);   // 256 MB ping
  float* bufB = alloc(67108864);   // 256 MB pong
  float* xyz0 = alloc(32 * 1024 * 3);
  float* h1 = alloc(2097152);
  float* h2 = alloc(2097152);
  float* f0 = alloc(2097152);
  float* nx0 = alloc(32 * 512 * 3);
  float* nx1 = alloc(32 * 256 * 3);
  int* fidx0 = (int*)alloc(32 * 512);
  int* nidx0 = (int*)alloc(32 * 512 * 32);
  int* fidx1 = (int*)alloc(32 * 256);
  int* nidx1 = (int*)alloc(32 * 256 * 32);
  float* sc = alloc(2048);
  float* shf = alloc(2048);

  // Sub-allocations inside the ping-pong buffers (valid after their producers die)
  float* Z = bufB;                 // (32,1280,256) concat buffer
  float* zz = bufA;                // (32,1024,256) fuse conv out
  float* P1 = bufA + 8388608;      // (32,256,256)
  float* Tb = P1 + 2097152;        // (32,256,256)
  float* Kb = Tb + 2097152;        // (32,64,256)
  float* Vb = Kb + 524288;         // (32,256,256)
  float* Att = Vb + 2097152;       // (32,256,256)
  float* xmr = Att + 2097152;      // (32,256,256)
  float* yb = xmr + 2097152;       // (32,256,256)
  float* raw = yb + 2097152;       // generic pre-BN conv output (32,C,N) <= 2M
  float* fvT = raw + 2097152;      // (1024,32)
  float* o1 = fvT + 32768;         // (512,32)
  float* o2 = o1 + 16384;          // (256,32)
  float* o3 = o2 + 8192;           // (40,32)
  float* cs = o3 + 2048;           // (32,256)

  auto gemm = [&](const float* A, const float* B, float* C, int M, int N, int K,
                  int lda, int ldb, int ldc, long sA, long sB, long sC, int batch,
                  const float* bias, const float* bSc, const float* bSh, bool transA) {
    dim3 g((N + 63) / 64, (M + 63) / 64, batch);
    if (transA)
      gemm_wmma<true><<<g, 128, 0, stream>>>(A, B, C, M, N, K, lda, ldb, ldc, sA, sB, sC, bias, bSc, bSh);
    else
      gemm_wmma<false><<<g, 128, 0, stream>>>(A, B, C, M, N, K, lda, ldb, ldc, sA, sB, sC, bias, bSc, bSh);
  };
  auto stats = [&](const float* x, int b, int C, long N, long bstride,
                   const float* g, const float* bt) {
    stats_kernel<<<C, 256, 0, stream>>>(x, b, C, N, bstride, g, bt, sc, shf);
  };
  auto ew = [&](const float* in, long is, const float* res, long rs, float* out,
                long os, int C, int N, int b, int act) {
    long total = (long)b * C * N;
    ew_affine<<<(int)((total + 255) / 256), 256, 0, stream>>>(in, is, res, rs, out, os, sc, shf, C, N, total, act);
  };

  // ---- stem: conv1(3->64)+BN+ReLU, conv2(64->64)+BN+ReLU ----
  gemm(P(1), X, raw, 64, 1024, 3, 3, 1024, 1024, 0, 3072, 65536, 32, nullptr, nullptr, nullptr, false);
  stats(raw, 32, 64, 1024, 65536, P(2), P(3));
  ew(raw, 65536, nullptr, 0, h1, 65536, 64, 1024, 32, 0);
  gemm(P(4), h1, raw, 64, 1024, 64, 64, 1024, 1024, 0, 65536, 65536, 32, nullptr, nullptr, nullptr, false);
  stats(raw, 32, 64, 1024, 65536, P(5), P(6));
  ew(raw, 65536, nullptr, 0, h2, 65536, 64, 1024, 32, 0);

  // ---- grouping stage 0: FPS(512) + kNN(32) + gather(2D=128) ----
  transpose_xyz<<<(32 * 1024 + 255) / 256, 256, 0, stream>>>(X, xyz0, 32, 1024);
  fps_kernel<<<32, 256, 0, stream>>>(xyz0, 1024, 512, fidx0, nx0);
  knn_kernel<<<dim3(512, 32), 128, 0, stream>>>(xyz0, nx0, 1024, 512, 32, nidx0);
  long G0 = 32L * 512 * 32;  // 524288 columns
  gather_feat<<<dim3((unsigned)((G0 + 255) / 256), 128), 256, 0, stream>>>(h2, nidx0, fidx0, bufA, 32, 64, 1024, 512, 32);
  // local_op g0: conv(128x128) -> BN stats -> conv(128x128) with fused BN+ReLU on B
  gemm(P(7), bufA, bufB, 128, (int)G0, 128, 128, (int)G0, (int)G0, 0, 0, 0, 1, nullptr, nullptr, nullptr, false);
  stats(bufB, 1, 128, G0, 0, P(8), P(9));
  gemm(P(10), bufB, bufA, 128, (int)G0, 128, 128, (int)G0, (int)G0, 0, 0, 0, 1, nullptr, sc, shf, false);
  stats(bufA, 1, 128, G0, 0, P(11), P(12));
  maxpool_local<<<(int)((32L * 128 * 512 + 255) / 256), 256, 0, stream>>>(bufA, sc, shf, f0, 128L * 512, 32, 128, 512, 32);

  // ---- grouping stage 1: FPS(256) + kNN(32) over new_xyz0, gather(2D=256) ----
  fps_kernel<<<32, 256, 0, stream>>>(nx0, 512, 256, fidx1, nx1);
  knn_kernel<<<dim3(256, 32), 128, 0, stream>>>(nx0, nx1, 512, 256, 32, nidx1);
  long G1 = 32L * 256 * 32;  // 262144 columns
  gather_feat<<<dim3((unsigned)((G1 + 255) / 256), 256), 256, 0, stream>>>(f0, nidx1, fidx1, bufA, 32, 128, 512, 256, 32);
  gemm(P(13), bufA, bufB, 256, (int)G1, 256, 256, (int)G1, (int)G1, 0, 0, 0, 1, nullptr, nullptr, nullptr, false);
  stats(bufB, 1, 256, G1, 0, P(14), P(15));
  gemm(P(16), bufB, bufA, 256, (int)G1, 256, 256, (int)G1, (int)G1, 0, 0, 0, 1, nullptr, sc, shf, false);
  stats(bufA, 1, 256, G1, 0, P(17), P(18));
  long sZ = 1280L * 256;
  float* f1 = Z + 1024 * 256;  // f1 lives directly in the concat buffer
  maxpool_local<<<(int)((32L * 256 * 256 + 255) / 256), 256, 0, stream>>>(bufA, sc, shf, f1, sZ, 32, 256, 256, 32);

  // ---- pt convs (256->256 twice, BN+ReLU) ----
  gemm(P(19), f1, raw, 256, 256, 256, 256, 256, 256, 0, sZ, 65536, 32, nullptr, nullptr, nullptr, false);
  stats(raw, 32, 256, 256, 65536, P(20), P(21));
  ew(raw, 65536, nullptr, 0, P1, 65536, 256, 256, 32, 0);
  gemm(P(22), P1, raw, 256, 256, 256, 256, 256, 256, 0, 65536, 65536, 32, nullptr, nullptr, nullptr, false);
  stats(raw, 32, 256, 256, 65536, P(23), P(24));
  ew(raw, 65536, nullptr, 0, Tb, 65536, 256, 256, 32, 0);

  // ---- 4 offset-attention layers; each writes its slice of Z ----
  for (int i = 0; i < 4; ++i) {
    int base = 25 + i * 7;
    const float* wqk = P(base + 0);
    const float* wv = P(base + 1);
    const float* bv = P(base + 2);
    const float* wt = P(base + 3);
    const float* bt = P(base + 4);
    const float* gg = P(base + 5);
    const float* bb = P(base + 6);
    const float* curp = (i == 0) ? Tb : (Z + (long)(i - 1) * 256 * 256);
    long curStride = (i == 0) ? 65536 : sZ;
    float* outp = Z + (long)i * 256 * 256;
    // K_ = wqk * cur (q shares weights)
    gemm(wqk, curp, Kb, 64, 256, 256, 256, 256, 256, 0, curStride, 64 * 256, 32, nullptr, nullptr, nullptr, false);
    // energy = K_^T * K_
    gemm(Kb, Kb, Att, 256, 256, 64, 256, 256, 256, 64 * 256, 64 * 256, 65536, 32, nullptr, nullptr, nullptr, true);
    softmax_rows<<<32 * 256, 256, 0, stream>>>(Att, 256);
    colsum_kernel<<<(32 * 256 + 255) / 256, 256, 0, stream>>>(Att, cs, 256, 32 * 256);
    renorm_kernel<<<(int)((32L * 65536 + 255) / 256), 256, 0, stream>>>(Att, cs, 256, 32L * 65536);
    // v = wv * cur + bv
    gemm(wv, curp, Vb, 256, 256, 256, 256, 256, 256, 0, curStride, 65536, 32, bv, nullptr, nullptr, false);
    // xr = v * att
    gemm(Vb, Att, yb, 256, 256, 256, 256, 256, 256, 65536, 65536, 65536, 32, nullptr, nullptr, nullptr, false);
    // xmr = cur - xr
    ew_sub<<<(int)((32L * 65536 + 255) / 256), 256, 0, stream>>>(curp, curStride, yb, 65536, xmr, 65536, 32L * 65536);
    // y = wt * xmr + bt ; out = cur + relu(BN(y))
    gemm(wt, xmr, raw, 256, 256, 256, 256, 256, 256, 0, 65536, 65536, 32, bt, nullptr, nullptr, false);
    stats(raw, 32, 256, 256, 65536, gg, bb);
    ew(raw, 65536, curp, curStride, outp, sZ, 256, 256, 32, 0);
  }

  // ---- fuse conv (1024x1280) + BN + leaky + maxpool ----
  gemm(P(53), Z, zz, 1024, 256, 1280, 1280, 256, 256, 0, sZ, 262144, 32, nullptr, nullptr, nullptr, false);
  stats(zz, 32, 1024, 256, 262144, P(54), P(55));
  maxpool_leaky_T<<<(1024 * 32 + 255) / 256, 256, 0, stream>>>(zz, sc, shf, fvT, 32, 1024, 256);

  // ---- MLP head ----
  gemm(P(56), fvT, o1, 512, 32, 1024, 1024, 32, 32, 0, 0, 0, 1, nullptr, nullptr, nullptr, false);
  stats(o1, 1, 512, 32, 0, P(57), P(58));
  ew(o1, 0, nullptr, 0, o1, 0, 512, 32, 1, 1);
  gemm(P(59), o1, o2, 256, 32, 512, 512, 32, 32, 0, 0, 0, 1, P(60), nullptr, nullptr, false);
  stats(o2, 1, 256, 32, 0, P(61), P(62));
  ew(o2, 0, nullptr, 0, o2, 0, 256, 32, 1, 1);
  gemm(P(63), o2, o3, 40, 32, 256, 256, 32, 32, 0, 0, 0, 1, P(64), nullptr, nullptr, false);
  final_store<<<(1280 + 255) / 256, 256, 0, stream>>>(o3, (float*)d_out, 32, 40);
}